// FuseUnit2_45148696216018
// MI455X (gfx1250) — compile-verified
//
#include <hip/hip_runtime.h>

typedef __attribute__((ext_vector_type(2))) float v2f;
typedef __attribute__((ext_vector_type(8))) float v8f;
typedef __attribute__((ext_vector_type(4))) int   v4i;

constexpr int IN_DIM  = 768;
constexpr int HID_DIM = 512;
constexpr int OUT_DIM = 256;
constexpr int TILE_B  = 16;               // batch rows per block (WMMA M)
constexpr int SPLIT   = 4;                // i (query) dimension split across blocks
constexpr int IPS     = HID_DIM / SPLIT;  // 128 i-columns per block
constexpr int LDT     = 516;              // padded LDS stride (516 mod 64 = 4 -> conflict-free)
constexpr int LDC     = IPS + 8;          // padded stride for the c slice

#if defined(__gfx1250__) && __has_builtin(__builtin_amdgcn_global_load_async_to_lds_b128)
#define HAVE_ASYNC_LDS 1
#else
#define HAVE_ASYNC_LDS 0
#endif

#if HAVE_ASYNC_LDS
typedef __attribute__((address_space(1))) v4i* gptr_b128;   // global int4*
typedef __attribute__((address_space(3))) v4i* lptr_b128;   // LDS int4*
#endif

__device__ __forceinline__ v8f wmma_f32_k4(v2f a, v2f b, v8f c) {
  // D = A(16x4, f32) * B(4x16, f32) + C   -> v_wmma_f32_16x16x4_f32
  return __builtin_amdgcn_wmma_f32_16x16x4_f32(false, a, false, b, (short)0, c,
                                               false, false);
}

// One 16x16 fp32 WMMA tile of Y = X @ W^T, K-unrolled by 4 (16 k-values per
// trip): 8 b64 loads in flight, then 4 back-to-back v_wmma ops.
__device__ __forceinline__ v8f gemm_tile(const float* __restrict__ xrow,
                                         const float* __restrict__ wrow,
                                         int K) {
  v8f acc = {};
  for (int k0 = 0; k0 < K; k0 += 16) {
    v2f a0 = *(const v2f*)(xrow + k0);
    v2f w0 = *(const v2f*)(wrow + k0);
    v2f a1 = *(const v2f*)(xrow + k0 + 4);
    v2f w1 = *(const v2f*)(wrow + k0 + 4);
    v2f a2 = *(const v2f*)(xrow + k0 + 8);
    v2f w2 = *(const v2f*)(wrow + k0 + 8);
    v2f a3 = *(const v2f*)(xrow + k0 + 12);
    v2f w3 = *(const v2f*)(wrow + k0 + 12);
    acc = wmma_f32_k4(a0, w0, acc);
    acc = wmma_f32_k4(a1, w1, acc);
    acc = wmma_f32_k4(a2, w2, acc);
    acc = wmma_f32_k4(a3, w3, acc);
  }
  return acc;
}

// Kernel 1: per (batch-tile, i-split) block:
//   t = s_emb @ W1^T + b1  (full 512 cols, LDS)
//   c = combin @ W2^T + b2 (this block's 128-col slice, LDS)
//   h[b,i] = relu( sum_j softmax_j(scale*c_i*t_j) * s_j )  -> global ws
__launch_bounds__(256)
__global__ void attn_h_kernel(const float* __restrict__ combin,  // [B, 768]
                              const float* __restrict__ s_emb,   // [B, 512]
                              const float* __restrict__ W1,      // [512, 512]
                              const float* __restrict__ b1,      // [512]
                              const float* __restrict__ W2,      // [512, 768]
                              const float* __restrict__ b2,      // [512]
                              float* __restrict__ h)             // [B, 512] (ws)
{
  __shared__ float t_s[TILE_B * LDT];
  __shared__ float s_s[TILE_B * LDT];
  __shared__ float c_s[TILE_B * LDC];
  __shared__ float pmax_s[TILE_B * 17];
  __shared__ float pmin_s[TILE_B * 17];
  __shared__ float tmax_s[TILE_B];
  __shared__ float tmin_s[TILE_B];

  const int tid   = threadIdx.x;
  const int lane  = tid & 31;
  const int wave  = tid >> 5;            // 8 waves
  const int lo    = lane & 15;
  const int hi    = lane >> 4;
  const int b0    = blockIdx.x * TILE_B;
  const int split = blockIdx.y;

  // ---- stage 0: raw s rows -> LDS ----
  for (int idx = tid; idx < TILE_B * (HID_DIM / 4); idx += 256) {
    const int r  = idx >> 7;             // / 128
    const int c4 = idx & 127;
    const float* src = s_emb + (size_t)(b0 + r) * HID_DIM + c4 * 4;
    float* dst = &s_s[r * LDT + c4 * 4];
#if HAVE_ASYNC_LDS
    // CDNA5 async DMA: GLOBAL_LOAD_ASYNC_TO_LDS_B128 (ASYNCcnt-tracked)
    __builtin_amdgcn_global_load_async_to_lds_b128(
        (gptr_b128)(unsigned long long)(uintptr_t)src,
        (lptr_b128)(unsigned)(uintptr_t)dst,
        0, 0);
#else
    const float4 v = *(const float4*)src;
    dst[0] = v.x; dst[1] = v.y; dst[2] = v.z; dst[3] = v.w;
#endif
  }

  // ---- stage 1: WMMA GEMMs, uniform per-wave work (no divergent selects) ----
  // t: 32 tiles, 4 per wave (fixed trip count)
#pragma unroll
  for (int rep = 0; rep < 4; ++rep) {
    const int n0 = (wave + rep * 8) * 16;
    const float* xrow = s_emb + (size_t)(b0 + lo) * HID_DIM + 2 * hi;
    const float* wrow = W1 + (size_t)(n0 + lo) * HID_DIM + 2 * hi;
    v8f acc = gemm_tile(xrow, wrow, HID_DIM);
    const float bb = b1[n0 + lo];
#pragma unroll
    for (int r = 0; r < 8; ++r)
      t_s[(r + 8 * hi) * LDT + n0 + lo] = acc[r] + bb;
  }
  // c slice: 8 tiles, exactly 1 per wave
  {
    const int n0 = split * IPS + wave * 16;
    const float* xrow = combin + (size_t)(b0 + lo) * IN_DIM + 2 * hi;
    const float* wrow = W2 + (size_t)(n0 + lo) * IN_DIM + 2 * hi;
    v8f acc = gemm_tile(xrow, wrow, IN_DIM);
    const float bb = b2[n0 + lo];
    const int cc = wave * 16 + lo;
#pragma unroll
    for (int r = 0; r < 8; ++r)
      c_s[(r + 8 * hi) * LDC + cc] = acc[r] + bb;
  }

#if HAVE_ASYNC_LDS
#if __has_builtin(__builtin_amdgcn_s_wait_asynccnt)
  __builtin_amdgcn_s_wait_asynccnt(0);
#else
  asm volatile("s_wait_asynccnt 0x0" ::: "memory");
#endif
#endif
  __syncthreads();

  // ---- per-row extrema of t: two-level parallel reduction ----
  {
    const int r  = tid >> 4;             // 16 threads per row
    const int sg = tid & 15;             // 32 elements per thread
    float mx = -3.402823466e38f, mn = 3.402823466e38f;
    const float* trp = &t_s[r * LDT + sg * 32];
#pragma unroll 4
    for (int j = 0; j < 32; ++j) {
      const float v = trp[j];
      mx = fmaxf(mx, v);
      mn = fminf(mn, v);
    }
    pmax_s[r * 17 + sg] = mx;
    pmin_s[r * 17 + sg] = mn;
  }
  __syncthreads();
  if (tid < TILE_B) {
    float mx = -3.402823466e38f, mn = 3.402823466e38f;
#pragma unroll
    for (int sg = 0; sg < 16; ++sg) {
      mx = fmaxf(mx, pmax_s[tid * 17 + sg]);
      mn = fminf(mn, pmin_s[tid * 17 + sg]);
    }
    tmax_s[tid] = mx;
    tmin_s[tid] = mn;
  }
  __syncthreads();

  // ---- stage 2: softmax-weighted reduction; exp is TRANS, co-executes
  //      with the fma accumulation. Lanes share (r, j) -> LDS broadcast. ----
  const float scale = 0.04419417382415922f;   // 1/sqrt(512)
  for (int it = 0; it < (TILE_B * IPS) / 256; ++it) {   // 8 iterations
    const int p  = it * 256 + tid;
    const int r  = p >> 7;                    // / IPS (=128)
    const int il = p & (IPS - 1);
    const float alpha = scale * c_s[r * LDC + il];
    const float m = alpha * (alpha >= 0.f ? tmax_s[r] : tmin_s[r]);
    const float* tr = &t_s[r * LDT];
    const float* sr = &s_s[r * LDT];
    float num = 0.f, den = 0.f;
#pragma unroll 4
    for (int j = 0; j < HID_DIM; ++j) {
      const float e = __expf(fmaf(alpha, tr[j], -m));
      den += e;
      num = fmaf(e, sr[j], num);
    }
    h[(size_t)(b0 + r) * HID_DIM + split * IPS + il] = fmaxf(num / den, 0.f);
  }
}

// Kernel 2: out = h @ W3^T + b3, one 16x16 WMMA tile per wave.
__launch_bounds__(256)
__global__ void gemm3_kernel(const float* __restrict__ h,    // [B, 512]
                             const float* __restrict__ W3,   // [256, 512]
                             const float* __restrict__ b3,   // [256]
                             float* __restrict__ out)        // [B, 256]
{
  const int tid  = threadIdx.x;
  const int lane = tid & 31;
  const int wave = tid >> 5;
  const int lo   = lane & 15;
  const int hi   = lane >> 4;

  const int tileid = blockIdx.x * 8 + wave;  // 1024 tiles = (1024/16) x (256/16)
  const int m0 = (tileid >> 4) * 16;         // batch tile
  const int n0 = (tileid & 15) * 16;         // out-col tile

  const float* arow = h  + (size_t)(m0 + lo) * HID_DIM + 2 * hi;
  const float* brow = W3 + (size_t)(n0 + lo) * HID_DIM + 2 * hi;
  v8f acc = gemm_tile(arow, brow, HID_DIM);

  const float bb = b3[n0 + lo];
#pragma unroll
  for (int r = 0; r < 8; ++r)
    out[(size_t)(m0 + r + 8 * hi) * OUT_DIM + n0 + lo] = acc[r] + bb;
}

extern "C" void kernel_launch(void* const* d_in, const int* in_sizes, int n_in,
                              void* d_out, int out_size, void* d_ws, size_t ws_size,
                              hipStream_t stream) {
  const float* combin = (const float*)d_in[0];
  const float* s_emb  = (const float*)d_in[1];
  const float* W1     = (const float*)d_in[2];
  const float* b1     = (const float*)d_in[3];
  const float* W2     = (const float*)d_in[4];
  const float* b2     = (const float*)d_in[5];
  const float* W3     = (const float*)d_in[6];
  const float* b3     = (const float*)d_in[7];
  float* out = (float*)d_out;

  const int B = in_sizes[0] / IN_DIM;        // 1024
  float* h = (float*)d_ws;                   // [B, 512] intermediate, 2 MB

  dim3 g1(B / TILE_B, SPLIT);                // (64, 4) = 256 workgroups
  attn_h_kernel<<<g1, 256, 0, stream>>>(combin, s_emb, W1, b1, W2, b2, h);

  const int tiles = (B / 16) * (OUT_DIM / 16);   // 1024 tiles
  gemm3_kernel<<<tiles / 8, 256, 0, stream>>>(h, W3, b3, out);

  (void)n_in; (void)out_size; (void)ws_size;
}